// SEBlock_11613591568561
// MI455X (gfx1250) — compile-verified
//
#include <hip/hip_runtime.h>
#include <math.h>

#define CH      256
#define BOT     32          // CH/8
#define BATCH   32
#define HW      3136        // 56*56
#define HW4     784         // HW/4 (float4 elements per plane)
#define NPLANES (BATCH * CH)   // 8192
#define NVEC4   (NPLANES * HW4) // 6,422,528 float4s total

typedef __attribute__((ext_vector_type(2))) float v2f;
typedef __attribute__((ext_vector_type(4))) float v4f;
typedef __attribute__((ext_vector_type(8))) float v8f;

// ---------------------------------------------------------------------------
// Kernel 1: global average pool over H,W.  One block per (b,c) plane.
// Regular-temporal loads deliberately warm the 192MB L2 with x (102.8MB fits)
// so the scale pass re-reads x from L2 instead of HBM.
// ---------------------------------------------------------------------------
__global__ __launch_bounds__(256) void se_pool(const float* __restrict__ x,
                                               float* __restrict__ s) {
    __shared__ float red[256];
    const int plane = blockIdx.x;
    const v4f* xp = reinterpret_cast<const v4f*>(x) + (size_t)plane * HW4;
    const int t = threadIdx.x;

    // 784 float4s per plane = 3*256 + 16
    v4f v0 = xp[t];
    v4f v1 = xp[t + 256];
    v4f v2 = xp[t + 512];
    float sum = v0.x + v0.y + v0.z + v0.w
              + v1.x + v1.y + v1.z + v1.w
              + v2.x + v2.y + v2.z + v2.w;
    if (t < (HW4 - 768)) {                 // 16-element tail
        v4f v3 = xp[t + 768];
        sum += v3.x + v3.y + v3.z + v3.w;
    }
    red[t] = sum;
    __syncthreads();
    for (int off = 128; off > 0; off >>= 1) {
        if (t < off) red[t] += red[t + off];
        __syncthreads();
    }
    if (t == 0) s[plane] = red[0] * (1.0f / (float)HW);
}

// ---------------------------------------------------------------------------
// Kernel 2: both FC layers with V_WMMA_F32_16X16X4_F32 (fp32 matrix path).
// One block of 8 wave32s.  All WMMA-executing branches are wave-uniform so
// EXEC is all-ones, as required by the ISA.
//
// Fragment layouts per CDNA5 ISA 7.12.2 (32-bit, 16x4 A / 4x16 B / 16x16 C):
//   A: lane L, row m = L%16;  v[0]=A[m][k+2*(L/16)],   v[1]=A[m][k+2*(L/16)+1]
//   B: lane L, col n = L%16;  v[0]=B[k+2*(L/16)][n],   v[1]=B[k+2*(L/16)+1][n]
//   C/D: lane L, col n = L%16; VGPR r = row (r + 8*(L/16))
// ---------------------------------------------------------------------------
__global__ __launch_bounds__(256) void se_fc(const float* __restrict__ s,
                                             const float* __restrict__ w1,
                                             const float* __restrict__ b1,
                                             const float* __restrict__ w2,
                                             const float* __restrict__ b2,
                                             float* __restrict__ g) {
    __shared__ float hbuf[BATCH * BOT];    // 32x32 hidden activations

    const int tid    = threadIdx.x;
    const int wave   = tid >> 5;
    const int lane   = tid & 31;
    const int lanelo = lane & 15;
    const int grp    = lane >> 4;          // 0 or 1 (half-wave)

    // ---- FC1: h = relu(s @ w1^T + b1)   M=32(batch) N=32(bot) K=256 ----
    if (wave < 4) {
        const int mt = wave >> 1, nt = wave & 1;
        const int m = mt * 16 + lanelo;    // batch row (A)
        const int n = nt * 16 + lanelo;    // bottleneck col (B): B[k][n] = w1[n][k]
        v8f acc = {};
        for (int k0 = 0; k0 < CH; k0 += 4) {
            const int ka = k0 + 2 * grp;
            v2f a = *reinterpret_cast<const v2f*>(s  + m * CH + ka);
            v2f b = *reinterpret_cast<const v2f*>(w1 + n * CH + ka);
            acc = __builtin_amdgcn_wmma_f32_16x16x4_f32(
                false, a, false, b, (short)0, acc, false, false);
        }
        const float bias = b1[n];
        #pragma unroll
        for (int r = 0; r < 8; ++r) {
            float hv = acc[r] + bias;
            hbuf[(mt * 16 + r + 8 * grp) * BOT + n] = hv > 0.0f ? hv : 0.0f;
        }
    }
    __syncthreads();

    // ---- FC2: g = sigmoid(h @ w2^T + b2)   M=32 N=256 K=32 ----
    // 32 output tiles (2 mtiles x 16 ntiles); 4 tiles per wave.
    for (int tt = 0; tt < 4; ++tt) {
        const int tile = wave * 4 + tt;
        const int mt = tile & 1;
        const int nt = tile >> 1;          // 0..15
        const int m = mt * 16 + lanelo;
        const int n = nt * 16 + lanelo;    // channel col: B[k][n] = w2[n][k]
        v8f acc = {};
        #pragma unroll
        for (int k0 = 0; k0 < BOT; k0 += 4) {
            const int ka = k0 + 2 * grp;
            v2f a = *reinterpret_cast<const v2f*>(&hbuf[m * BOT + ka]);  // ds b64
            v2f b = *reinterpret_cast<const v2f*>(w2 + n * BOT + ka);
            acc = __builtin_amdgcn_wmma_f32_16x16x4_f32(
                false, a, false, b, (short)0, acc, false, false);
        }
        const float bias = b2[n];
        #pragma unroll
        for (int r = 0; r < 8; ++r) {
            const float z  = acc[r] + bias;
            const float gv = 1.0f / (1.0f + __expf(-z));
            g[(mt * 16 + r + 8 * grp) * CH + n] = gv;
        }
    }
}

// ---------------------------------------------------------------------------
// Kernel 3: out = x * g[b,c].  x loads hit L2 (warmed by se_pool); out is
// streamed with non-temporal stores so it doesn't evict x / pollute L2.
// ---------------------------------------------------------------------------
__global__ __launch_bounds__(256) void se_scale(const float* __restrict__ x,
                                                const float* __restrict__ g,
                                                float* __restrict__ out) {
    const int i = blockIdx.x * 256 + threadIdx.x;   // float4 index, exact grid
    const int plane = i / HW4;                      // const-div -> mul/shift
    const float gv = g[plane];
    v4f v = reinterpret_cast<const v4f*>(x)[i];
    v4f o = { v.x * gv, v.y * gv, v.z * gv, v.w * gv };
    __builtin_nontemporal_store(o, reinterpret_cast<v4f*>(out) + i);
}

// ---------------------------------------------------------------------------
extern "C" void kernel_launch(void* const* d_in, const int* in_sizes, int n_in,
                              void* d_out, int out_size, void* d_ws, size_t ws_size,
                              hipStream_t stream) {
    const float* x  = (const float*)d_in[0];   // [32,256,56,56]
    const float* w1 = (const float*)d_in[1];   // [32,256]
    const float* b1 = (const float*)d_in[2];   // [32]
    const float* w2 = (const float*)d_in[3];   // [256,32]
    const float* b2 = (const float*)d_in[4];   // [256]
    float* out = (float*)d_out;

    float* s = (float*)d_ws;                   // [32*256] pooled means
    float* g = s + NPLANES;                    // [32*256] gates

    se_pool<<<NPLANES, 256, 0, stream>>>(x, s);
    se_fc<<<1, 256, 0, stream>>>(s, w1, b1, w2, b2, g);
    se_scale<<<NVEC4 / 256, 256, 0, stream>>>(x, g, out);
}